// Torch_LRU_16853451670161
// MI455X (gfx1250) — compile-verified
//
#include <hip/hip_runtime.h>
#include <hip/hip_bf16.h>

typedef float v2f __attribute__((ext_vector_type(2)));
typedef float v8f __attribute__((ext_vector_type(8)));
typedef int   i32x2 __attribute__((vector_size(8)));
typedef int   i32x4 __attribute__((vector_size(16)));

#define N_ROWS    32768
#define IN_DIM    256
#define STATE_DIM 512
#define OUT_DIM   256

#define WPB      4      // waves per block (4 x 16 = 64 rows per block)
#define USTRIDE  260    // u-tile LDS stride (floats): 16B-aligned rows, conflict-free frags
#define SSTRIDE  18     // s-tile LDS stride (floats)
#define CSTRIDE  272    // C/D slice LDS stride (float2): +pair lands 32 banks away

// ---- CDNA5 async global->LDS copies (guarded; fall back to plain ld/st) ----
#if __has_builtin(__builtin_amdgcn_global_load_async_to_lds_b64) && \
    __has_builtin(__builtin_amdgcn_global_load_async_to_lds_b128) && \
    __has_builtin(__builtin_amdgcn_s_wait_asynccnt)
#define USE_ASYNC 1
#else
#define USE_ASYNC 0
#endif

__device__ __forceinline__ void cp_b64(const void* gsrc, void* ldst) {
#if USE_ASYNC
    __builtin_amdgcn_global_load_async_to_lds_b64(
        (__attribute__((address_space(1))) i32x2*)(uintptr_t)gsrc,
        (__attribute__((address_space(3))) i32x2*)(unsigned int)(uintptr_t)ldst, 0, 0);
#else
    *(float2*)ldst = *(const float2*)gsrc;
#endif
}
__device__ __forceinline__ void cp_b128(const void* gsrc, void* ldst) {
#if USE_ASYNC
    __builtin_amdgcn_global_load_async_to_lds_b128(
        (__attribute__((address_space(1))) i32x4*)(uintptr_t)gsrc,
        (__attribute__((address_space(3))) i32x4*)(unsigned int)(uintptr_t)ldst, 0, 0);
#else
    float4 v = *(const float4*)gsrc;
    float* d = (float*)ldst;
    d[0] = v.x; d[1] = v.y; d[2] = v.z; d[3] = v.w;
#endif
}
__device__ __forceinline__ void wait_async() {
#if USE_ASYNC
    __builtin_amdgcn_s_wait_asynccnt(0);
#endif
}

// ---------------------------------------------------------------------------
// Prologue: compute lambda, pre-pack weights in WMMA-B-fragment order.
// Bp[kpair][n] = {W[2k][n], W[2k+1][n]} -> one coalesced float2 per B frag.
// ---------------------------------------------------------------------------
__global__ void lru_prep(const float* __restrict__ nu_log,
                         const float* __restrict__ theta_log,
                         const float* __restrict__ gamma_log,
                         const float* __restrict__ B_re, const float* __restrict__ B_im,
                         const float* __restrict__ C_re, const float* __restrict__ C_im,
                         const float* __restrict__ D,
                         float* __restrict__ lam_re, float* __restrict__ lam_im,
                         float2* __restrict__ Bp_re, float2* __restrict__ Bp_im,
                         float2* __restrict__ Cp_re, float2* __restrict__ Cp_imn,
                         float2* __restrict__ Dp)
{
    const int tid = blockIdx.x * blockDim.x + threadIdx.x;
    const int stride = gridDim.x * blockDim.x;

    for (int i = tid; i < STATE_DIM; i += stride) {
        float lm = expf(-expf(nu_log[i]));
        float th = expf(theta_log[i]);
        lam_re[i] = lm * cosf(th);
        lam_im[i] = lm * sinf(th);
    }
    for (int i = tid; i < 128 * STATE_DIM; i += stride) {   // Bn^T packed
        int kp = i >> 9, h = i & 511;
        float g = expf(gamma_log[h]);
        const float* br = B_re + h * IN_DIM + 2 * kp;
        const float* bi = B_im + h * IN_DIM + 2 * kp;
        Bp_re[kp * STATE_DIM + h] = make_float2(br[0] * g, br[1] * g);
        Bp_im[kp * STATE_DIM + h] = make_float2(bi[0] * g, bi[1] * g);
    }
    for (int i = tid; i < 256 * OUT_DIM; i += stride) {     // C^T packed, C_im negated
        int kp = i >> 8, o = i & 255;
        const float* cr = C_re + (size_t)o * STATE_DIM + 2 * kp;
        const float* ci = C_im + (size_t)o * STATE_DIM + 2 * kp;
        Cp_re[kp * OUT_DIM + o]  = make_float2(cr[0], cr[1]);
        Cp_imn[kp * OUT_DIM + o] = make_float2(-ci[0], -ci[1]);
    }
    for (int i = tid; i < 128 * OUT_DIM; i += stride) {     // D^T packed
        int kp = i >> 8, o = i & 255;
        const float* dp = D + (size_t)o * IN_DIM + 2 * kp;
        Dp[kp * OUT_DIM + o] = make_float2(dp[0], dp[1]);
    }
}

// ---------------------------------------------------------------------------
// Main fused kernel: 4 waves/block, one 16-row tile per wave; weight slices
// staged cooperatively in LDS (async-to-LDS) and shared by all waves.
// ---------------------------------------------------------------------------
__global__ __launch_bounds__(WPB * 32) void lru_main(
    const float* __restrict__ u,
    const float* __restrict__ x0_re, const float* __restrict__ x0_im,
    const float* __restrict__ lam_re, const float* __restrict__ lam_im,
    const float2* __restrict__ Bp_re, const float2* __restrict__ Bp_im,
    const float2* __restrict__ Cp_re, const float2* __restrict__ Cp_imn,
    const float2* __restrict__ Dp,
    float* __restrict__ y_out, float2* __restrict__ state_out)
{
    __shared__ float  uT[WPB][16 * USTRIDE];        // 66.6 KB
    __shared__ float  sRe[WPB][16 * SSTRIDE];       // 4.6 KB
    __shared__ float  sIm[WPB][16 * SSTRIDE];       // 4.6 KB
    __shared__ float2 wPool[16 * CSTRIDE];          // 34.8 KB: B slice (re|im) or D chunk
    __shared__ float2 wCre[8 * CSTRIDE];            // 17.4 KB
    __shared__ float2 wCim[8 * CSTRIDE];            // 17.4 KB

    const int tid  = threadIdx.x;
    const int lane = tid & 31;
    const int wid  = tid >> 5;
    const int tile = blockIdx.x * WPB + wid;
    const long r0  = (long)tile * 16;

    float* ut  = uT[wid];
    float* sre = sRe[wid];
    float* sim = sIm[wid];
    float2* wBre = wPool;           // [kp 0..127][n 0..15]
    float2* wBim = wPool + 2048;

    const int m16 = lane & 15;      // M (or N) index within 16
    const int hi  = lane >> 4;      // selects K pair {0,1} vs {2,3}
    const int kof = hi * 2;

    // ---- stage u tile (16x256 contiguous) into LDS (async b128) ----
    {
        const float* usrc = u + r0 * IN_DIM;
        for (int i = lane; i < 16 * 64; i += 32) {
            int row = i >> 6;
            int col = (i & 63) * 4;
            cp_b128(usrc + (size_t)i * 4, ut + row * USTRIDE + col);
        }
    }

    v8f yacc[16];
#pragma unroll
    for (int ot = 0; ot < 16; ++ot)
        yacc[ot] = (v8f){0.f, 0.f, 0.f, 0.f, 0.f, 0.f, 0.f, 0.f};

#pragma unroll 1
    for (int hb = 0; hb < STATE_DIM / 16; ++hb) {
        const int h0 = hb * 16;

        __syncthreads();                       // previous slice consumers done
        // stage B slice: 128 kpairs x 16 h, re+im
        for (int idx = tid; idx < 2048; idx += WPB * 32) {
            int kp = idx >> 4, n = idx & 15;
            cp_b64(&Bp_re[kp * STATE_DIM + h0 + n], &wBre[kp * 16 + n]);
            cp_b64(&Bp_im[kp * STATE_DIM + h0 + n], &wBim[kp * 16 + n]);
        }
        // stage C slice: 8 kpairs x 256 o, re+(-im)
        for (int idx = tid; idx < 2048; idx += WPB * 32) {
            int cp = idx >> 8, o = idx & 255;
            cp_b64(&Cp_re[(h0 / 2 + cp) * OUT_DIM + o],  &wCre[cp * CSTRIDE + o]);
            cp_b64(&Cp_imn[(h0 / 2 + cp) * OUT_DIM + o], &wCim[cp * CSTRIDE + o]);
        }
        wait_async();
        __syncthreads();                       // slice visible to all waves

        // ---- Bu tile over K = 256 ----
        v8f accR = (v8f){0.f, 0.f, 0.f, 0.f, 0.f, 0.f, 0.f, 0.f};
        v8f accI = (v8f){0.f, 0.f, 0.f, 0.f, 0.f, 0.f, 0.f, 0.f};
#pragma unroll 4
        for (int ks = 0; ks < IN_DIM / 4; ++ks) {
            v2f a = *(const v2f*)(ut + m16 * USTRIDE + ks * 4 + kof);
            float2 br2 = wBre[(ks * 2 + hi) * 16 + m16];
            float2 bi2 = wBim[(ks * 2 + hi) * 16 + m16];
            v2f br = (v2f){br2.x, br2.y};
            v2f bi = (v2f){bi2.x, bi2.y};
            accR = __builtin_amdgcn_wmma_f32_16x16x4_f32(false, a, false, br,
                                                         (short)0, accR, false, false);
            accI = __builtin_amdgcn_wmma_f32_16x16x4_f32(false, a, false, bi,
                                                         (short)0, accI, false, false);
        }

        // ---- s = lam*x0 + Bu ; state out ; stage s tile in LDS ----
        const float lr = lam_re[h0 + m16];
        const float li = lam_im[h0 + m16];
#pragma unroll
        for (int v = 0; v < 8; ++v) {
            int  m   = v + 8 * hi;
            long row = r0 + m;
            int  h   = h0 + m16;
            float xr = x0_re[row * STATE_DIM + h];
            float xi = x0_im[row * STATE_DIM + h];
            float srv = fmaf(lr, xr, fmaf(-li, xi, accR[v]));
            float siv = fmaf(lr, xi, fmaf(li, xr, accI[v]));
            state_out[row * STATE_DIM + h] = make_float2(srv, siv);
            sre[m * SSTRIDE + m16] = srv;
            sim[m * SSTRIDE + m16] = siv;
        }

        // ---- y += s_re @ C_re^T + s_im @ (-C_im^T) over this 16-wide K ----
#pragma unroll
        for (int kk = 0; kk < 4; ++kk) {
            v2f ar = *(const v2f*)(sre + m16 * SSTRIDE + kk * 4 + kof);
            v2f ai = *(const v2f*)(sim + m16 * SSTRIDE + kk * 4 + kof);
            int cp = kk * 2 + hi;
#pragma unroll
            for (int ot = 0; ot < 16; ++ot) {
                float2 cr2 = wCre[cp * CSTRIDE + ot * 16 + m16];
                float2 ci2 = wCim[cp * CSTRIDE + ot * 16 + m16];
                v2f cr = (v2f){cr2.x, cr2.y};
                v2f ci = (v2f){ci2.x, ci2.y};
                yacc[ot] = __builtin_amdgcn_wmma_f32_16x16x4_f32(false, ar, false, cr,
                                                                 (short)0, yacc[ot], false, false);
                yacc[ot] = __builtin_amdgcn_wmma_f32_16x16x4_f32(false, ai, false, ci,
                                                                 (short)0, yacc[ot], false, false);
            }
        }
    }

    // ---- y += u @ D^T, staged in 8 chunks of 16 kpairs ----
#pragma unroll 1
    for (int cs = 0; cs < 8; ++cs) {
        __syncthreads();
        for (int idx = tid; idx < 4096; idx += WPB * 32) {
            int p = idx >> 8, o = idx & 255;
            cp_b64(&Dp[(cs * 16 + p) * OUT_DIM + o], &wPool[p * CSTRIDE + o]);
        }
        wait_async();
        __syncthreads();
#pragma unroll 2
        for (int k2 = 0; k2 < 8; ++k2) {
            int ks = cs * 8 + k2;
            v2f a = *(const v2f*)(ut + m16 * USTRIDE + ks * 4 + kof);
            int p = k2 * 2 + hi;
#pragma unroll
            for (int ot = 0; ot < 16; ++ot) {
                float2 d2 = wPool[p * CSTRIDE + ot * 16 + m16];
                v2f b = (v2f){d2.x, d2.y};
                yacc[ot] = __builtin_amdgcn_wmma_f32_16x16x4_f32(false, a, false, b,
                                                                 (short)0, yacc[ot], false, false);
            }
        }
    }

    // ---- store y (coalesced) ----
#pragma unroll
    for (int ot = 0; ot < 16; ++ot) {
#pragma unroll
        for (int v = 0; v < 8; ++v) {
            long row = r0 + v + 8 * hi;
            y_out[row * OUT_DIM + ot * 16 + m16] = yacc[ot][v];
        }
    }
}

// ---------------------------------------------------------------------------
extern "C" void kernel_launch(void* const* d_in, const int* in_sizes, int n_in,
                              void* d_out, int out_size, void* d_ws, size_t ws_size,
                              hipStream_t stream) {
    const float* u         = (const float*)d_in[0];
    const float* x0_re     = (const float*)d_in[1];
    const float* x0_im     = (const float*)d_in[2];
    const float* nu_log    = (const float*)d_in[3];
    const float* theta_log = (const float*)d_in[4];
    const float* gamma_log = (const float*)d_in[5];
    const float* B_re      = (const float*)d_in[6];
    const float* B_im      = (const float*)d_in[7];
    const float* C_re      = (const float*)d_in[8];
    const float* C_im      = (const float*)d_in[9];
    const float* D         = (const float*)d_in[10];

    float*  ws     = (float*)d_ws;
    float*  lam_re = ws;                        // 512
    float*  lam_im = ws + 512;                  // 512
    float2* Bp_re  = (float2*)(ws + 1024);      // 128*512 float2
    float2* Bp_im  = Bp_re + 128 * STATE_DIM;
    float2* Cp_re  = Bp_im + 128 * STATE_DIM;   // 256*256 float2
    float2* Cp_imn = Cp_re + 256 * OUT_DIM;
    float2* Dp     = Cp_imn + 256 * OUT_DIM;    // 128*256 float2

    float*  y_out     = (float*)d_out;
    float2* state_out = (float2*)((float*)d_out + (size_t)N_ROWS * OUT_DIM);

    lru_prep<<<512, 256, 0, stream>>>(nu_log, theta_log, gamma_log,
                                      B_re, B_im, C_re, C_im, D,
                                      lam_re, lam_im, Bp_re, Bp_im, Cp_re, Cp_imn, Dp);

    const int tiles  = N_ROWS / 16;             // 2048 row tiles
    const int blocks = tiles / WPB;             // 512 blocks of 4 waves
    lru_main<<<blocks, WPB * 32, 0, stream>>>(u, x0_re, x0_im, lam_re, lam_im,
                                              Bp_re, Bp_im, Cp_re, Cp_imn, Dp,
                                              y_out, state_out);
}